// DepthMask2PointCloudFast_53472342835417
// MI455X (gfx1250) — compile-verified
//
#include <hip/hip_runtime.h>
#include <cmath>

#define HH   150
#define WW   200
#define HW   (HH * WW)                // 30000
#define PCAP 1024
#define MPER 5
#define NT   256                      // 8 wave32 waves
#define CSTRIDE (MPER * (PCAP + 1))   // 5125
#define SENT 0xFFFFFFFFu              // "not masked" sentinel (valid keys have sign bit 0)

// ---- gfx1250 async global->LDS copy (guarded; falls back to plain loads) ----
#if defined(__has_builtin)
# if __has_builtin(__builtin_amdgcn_global_load_async_to_lds_b32)
#  define HAVE_ASYNC_COPY 1
# else
#  define HAVE_ASYNC_COPY 0
# endif
# if __has_builtin(__builtin_amdgcn_s_wait_asynccnt)
#  define HAVE_WAIT_ASYNC 1
# else
#  define HAVE_WAIT_ASYNC 0
# endif
#else
# define HAVE_ASYNC_COPY 0
# define HAVE_WAIT_ASYNC 0
#endif

#if HAVE_ASYNC_COPY
typedef __attribute__((address_space(1))) int gas_int;   // global (AS1)
typedef __attribute__((address_space(3))) int lds_int;   // LDS (AS3)
#endif

__device__ __forceinline__ void async_copy_f32(const float* g, float* l)
{
#if HAVE_ASYNC_COPY
    __builtin_amdgcn_global_load_async_to_lds_b32(
        (gas_int*)(void*)g, (lds_int*)(void*)l, 0, 0);
#else
    *l = *g;
#endif
}

__device__ __forceinline__ void async_wait_le1()
{
#if HAVE_ASYNC_COPY
# if HAVE_WAIT_ASYNC
    __builtin_amdgcn_s_wait_asynccnt(1);
# else
    asm volatile("s_wait_asynccnt 0x1" ::: "memory");
# endif
#endif
}

__device__ __forceinline__ void async_wait_le0()
{
#if HAVE_ASYNC_COPY
# if HAVE_WAIT_ASYNC
    __builtin_amdgcn_s_wait_asynccnt(0);
# else
    asm volatile("s_wait_asynccnt 0x0" ::: "memory");
# endif
#endif
}

// Parallel inclusive/exclusive scan of a 256-bin LDS histogram (bin = tid).
// Returns this thread's bin [excl, incl) range and the grand total.
__device__ __forceinline__ void scan_hist_256(const unsigned* hist, unsigned* psum, int tid,
                                              unsigned& excl, unsigned& incl, unsigned& total)
{
    unsigned v = hist[tid];
    psum[tid] = v;
    __syncthreads();
#pragma unroll
    for (int o = 1; o < 256; o <<= 1) {
        unsigned t = (tid >= o) ? psum[tid - o] : 0u;
        __syncthreads();
        psum[tid] += t;
        __syncthreads();
    }
    incl  = psum[tid];
    excl  = incl - v;
    total = psum[255];
}

__global__ __launch_bounds__(NT)
void DepthMask2PointCloud_kernel(const float* __restrict__ in,
                                 float* __restrict__ out,
                                 double fx_d, double fy_d)
{
    const int pidx = blockIdx.x;          // person 0..4  -> p = pidx+1
    const int b    = blockIdx.y;          // batch
    const int tid  = threadIdx.x;
    const int wave = tid >> 5;
    const int lane = tid & 31;
    const float pf = (float)(pidx + 1);

    const float* __restrict__ depth = in + (size_t)b * 3 * HW;      // channel 0
    const float* __restrict__ ind   = depth + HW;                   // channel 1

    float* __restrict__ outb = out + (size_t)b * 3 * CSTRIDE + (size_t)pidx * (PCAP + 1);

    // ---- LDS ----
    __shared__ alignas(16) unsigned s_keys[HW];   // 120000 B
    __shared__ unsigned hist[256];
    __shared__ unsigned s_psum[256];
    __shared__ float    s_stage[2][NT];           // double-buffered ind staging
    __shared__ unsigned s_pref[4];
    __shared__ unsigned s_rank[4];
    __shared__ unsigned s_wavecnt[NT / 32];
    __shared__ unsigned s_running;

    // ---- zero-fill this task's output region: 3 rows x (P+1) ----
    for (int j = tid; j < 3 * (PCAP + 1); j += NT) {
        int c = j / (PCAP + 1);
        int k = j - c * (PCAP + 1);
        outb[(size_t)c * CSTRIDE + k] = 0.0f;
    }

    hist[tid] = 0;
    if (tid == 0) s_running = 0;
    __syncthreads();

    // ---- pass 0 (only global pass): build keys in LDS + MSB histogram ----
    // Pipeline: async-stage ind chunk c+1 to LDS while loading depth chunk c.
    const int nchunks = (HW + NT - 1) / NT;       // 118
    {
        int i0 = tid;
        if (i0 < HW) async_copy_f32(ind + i0, &s_stage[0][tid]);
    }
    for (int c = 0; c < nchunks; ++c) {
        const int i = c * NT + tid;
        if (c + 1 < nchunks) {
            int j = (c + 1) * NT + tid;
            if (j < HW) async_copy_f32(ind + j, &s_stage[(c + 1) & 1][tid]);
            async_wait_le1();                     // chunk c staged for this wave
        } else {
            async_wait_le0();
        }
        if (i < HW) {
            __builtin_prefetch(depth + i + 4 * NT, 0, 0);
            float d  = depth[i];
            float iv = s_stage[c & 1][tid];       // own lane's staged copy
            unsigned key = SENT;
            if (rintf(iv) == pf && d > 3.0f) {
                key = __float_as_uint(d);         // positive -> monotone bits
                atomicAdd(&hist[key >> 24], 1u);
            }
            s_keys[i] = key;
        }
    }
    __syncthreads();

    // ---- resolve MSB byte for all 4 targets via parallel histogram scan ----
    unsigned excl, incl, total;
    scan_hist_256(hist, s_psum, tid, excl, incl, total);

    const unsigned n = total;                     // same in every thread
    float frac1 = 0.0f, frac3 = 0.0f;
    unsigned rk[4] = {0, 0, 0, 0};
    if (n > 0) {
        float nf    = (float)n;
        float basef = fmaxf(nf - 1.0f, 0.0f);
        float pos1  = 0.25f * basef;
        float pos3  = 0.75f * basef;
        int lo1 = (int)floorf(pos1), hi1 = (int)ceilf(pos1);
        int lo3 = (int)floorf(pos3), hi3 = (int)ceilf(pos3);
        frac1 = pos1 - (float)lo1;
        frac3 = pos3 - (float)lo3;
        rk[0] = (unsigned)lo1; rk[1] = (unsigned)hi1;
        rk[2] = (unsigned)lo3; rk[3] = (unsigned)hi3;
#pragma unroll
        for (int u = 0; u < 4; ++u) {
            if (rk[u] >= excl && rk[u] < incl) {  // exactly one thread matches
                s_pref[u] = (unsigned)tid << 24;
                s_rank[u] = rk[u] - excl;
            }
        }
    }
    __syncthreads();

    // ---- passes 1..3: byte refinement, entirely from LDS keys ----
    float lb, ub;
    if (n > 0) {
        for (int pass = 1; pass <= 3; ++pass) {
            const int shift = 24 - 8 * pass;
            const unsigned prefMask = 0xFFFFFFFFu << (32 - 8 * pass);

            unsigned pre[4];
#pragma unroll
            for (int t = 0; t < 4; ++t) pre[t] = s_pref[t];   // snapshot

            for (int t = 0; t < 4; ++t) {
                bool dup = false;
                for (int t2 = 0; t2 < t; ++t2) if (pre[t2] == pre[t]) { dup = true; break; }
                if (dup) continue;

                hist[tid] = 0;
                __syncthreads();

                const unsigned pref = pre[t];
                for (int i = tid; i < HW; i += NT) {
                    unsigned key = s_keys[i];
                    if (key != SENT && (key & prefMask) == pref)
                        atomicAdd(&hist[(key >> shift) & 0xFF], 1u);
                }
                __syncthreads();

                unsigned gexcl, gincl, gtotal;
                scan_hist_256(hist, s_psum, tid, gexcl, gincl, gtotal);

                unsigned rr[4];
#pragma unroll
                for (int u = 0; u < 4; ++u) rr[u] = s_rank[u];
                __syncthreads();                  // all reads done before writes

                for (int u = t; u < 4; ++u) {
                    if (pre[u] != pref) continue;
                    if (rr[u] >= gexcl && rr[u] < gincl) {
                        s_pref[u] = pref | ((unsigned)tid << shift);
                        s_rank[u] = rr[u] - gexcl;
                    }
                }
                __syncthreads();
            }
        }

        // every thread computes fences redundantly in registers
        float v0 = __uint_as_float(s_pref[0]);
        float v1 = __uint_as_float(s_pref[1]);
        float v2 = __uint_as_float(s_pref[2]);
        float v3 = __uint_as_float(s_pref[3]);
        float q1 = v0 * (1.0f - frac1) + v1 * frac1;
        float q3 = v2 * (1.0f - frac3) + v3 * frac3;
        float iqr = q3 - q1;
        lb = q1 - 1.5f * iqr;
        ub = q3 + 1.5f * iqr;
    } else {
        lb = __int_as_float(0x7fffffff);          // NaN fences -> mask2 all false
        ub = __int_as_float(0x7fffffff);
    }

    // ---- pass B: stable raster-order compaction, 4 keys/thread + wave32 scan ----
    for (int base = 0; base < HW; base += NT * 4) {
        unsigned base_run = s_running;
        if (base_run >= PCAP) break;              // uniform: flag & slots decided

        const int idx0 = base + tid * 4;
        unsigned k4[4];
        if (idx0 + 3 < HW) {
            uint4 v = *(const uint4*)&s_keys[idx0];   // ds_load_b128 (16B aligned)
            k4[0] = v.x; k4[1] = v.y; k4[2] = v.z; k4[3] = v.w;
        } else {
#pragma unroll
            for (int e = 0; e < 4; ++e)
                k4[e] = (idx0 + e < HW) ? s_keys[idx0 + e] : SENT;
        }

        bool  valid[4];
        float dv[4];
        unsigned cnt = 0;
#pragma unroll
        for (int e = 0; e < 4; ++e) {
            float d = __uint_as_float(k4[e]);
            bool v = (k4[e] != SENT) && (d >= lb) && (d <= ub);   // NaN -> false
            valid[e] = v;
            dv[e] = d;
            cnt += v ? 1u : 0u;
        }

        // wave32 inclusive scan of per-thread counts
        unsigned pre = cnt;
#pragma unroll
        for (int o = 1; o < 32; o <<= 1) {
            unsigned t = __shfl_up(pre, o, 32);
            if (lane >= o) pre += t;
        }
        unsigned laneExcl = pre - cnt;

        if (lane == 31) s_wavecnt[wave] = pre;    // wave total
        __syncthreads();

        unsigned off = 0, total2 = 0;
#pragma unroll
        for (int w = 0; w < NT / 32; ++w) {
            unsigned cc = s_wavecnt[w];
            if (w < wave) off += cc;
            total2 += cc;
        }

        unsigned rank = base_run + off + laneExcl;
#pragma unroll
        for (int e = 0; e < 4; ++e) {
            if (valid[e]) {
                if (rank < PCAP) {
                    int idx = idx0 + e;
                    int row = idx / WW;
                    int col = idx - row * WW;
                    float xc = (float)(((double)col - (double)WW / 2.0) / fx_d);
                    float yc = (float)(((double)row - (double)HH / 2.0) / fy_d);
                    outb[rank]                       = xc * dv[e];
                    outb[(size_t)CSTRIDE + rank]     = yc * dv[e];
                    outb[(size_t)2 * CSTRIDE + rank] = dv[e];
                }
                ++rank;
            }
        }

        if (tid == 0) s_running = base_run + total2;
        __syncthreads();
    }

    if (tid == 0) {
        outb[PCAP] = (s_running > 0) ? 1.0f : 0.0f;   // flag column (c=1,2 stay 0)
    }
}

extern "C" void kernel_launch(void* const* d_in, const int* in_sizes, int n_in,
                              void* d_out, int out_size, void* d_ws, size_t ws_size,
                              hipStream_t stream)
{
    (void)n_in; (void)d_ws; (void)ws_size; (void)out_size;
    const float* in = (const float*)d_in[0];
    float* out = (float*)d_out;

    const int B = in_sizes[0] / (3 * HW);   // 256

    // Match numpy float64 camera intrinsics exactly.
    const double PI = 3.14159265358979323846;
    const double fx = (double)WW / (2.0 * std::tan((81.0 * PI / 180.0) / 2.0));
    const double fy = (double)HH / (2.0 * std::tan((59.0 * PI / 180.0) / 2.0));

    dim3 grid(MPER, B);
    DepthMask2PointCloud_kernel<<<grid, NT, 0, stream>>>(in, out, fx, fy);
}